// standalone_self_attention_2D_48189533061167
// MI455X (gfx1250) — compile-verified
//
#include <hip/hip_runtime.h>
#include <hip/hip_bf16.h>

typedef __attribute__((ext_vector_type(2))) float v2f;
typedef __attribute__((ext_vector_type(8))) float v8f;

constexpr int Bc   = 4;
constexpr int Hc   = 512;
constexpr int Wc   = 229;
constexpr int Cc   = 32;
constexpr int OUTc = 88;
constexpr int Mrows = Bc * Hc;          // 2048 GEMM rows
constexpr int KPAD  = 232;              // 229 padded to mult. of 4
constexpr int NPAD  = 96;               // 88 padded to mult. of 16
constexpr int WLIN_STRIDE = Cc * Wc;    // 7328
constexpr int FP_SIZE = Bc * Hc * OUTc; // 180224 floats (frame_pred)

// workspace layout (in floats)
constexpr int WS_SCAL  = 0;                       // [0]=alpha, [1..3]=rt, [4..6]=rf
constexpr int WS_WEFFT = 32;                      // WeffT[KPAD][NPAD]
constexpr int WS_OV    = WS_WEFFT + KPAD * NPAD;  // ov[Mrows][KPAD]

// ---------------------------------------------------------------------------
// Kernel 0: alpha = Wq.Wk ; rt[i] = sum_{c<16} Wq[c]*rel_t[c,i] ;
//           rf[j] = sum_{c<16} Wq[16+c]*rel_f[c,j]
// ---------------------------------------------------------------------------
__global__ void __launch_bounds__(32)
k_scalars(const float* __restrict__ Wq, const float* __restrict__ Wk,
          const float* __restrict__ rel_t, const float* __restrict__ rel_f,
          float* __restrict__ scal) {
  int t = threadIdx.x;
  if (t == 0) {
    float a = 0.f;
    for (int c = 0; c < Cc; ++c) a += Wq[c] * Wk[c];
    scal[0] = a;
  } else if (t <= 3) {
    int i = t - 1;
    float a = 0.f;
    for (int c = 0; c < 16; ++c) a += Wq[c] * rel_t[c * 3 + i];
    scal[t] = a;
  } else if (t <= 6) {
    int j = t - 4;
    float a = 0.f;
    for (int c = 0; c < 16; ++c) a += Wq[16 + c] * rel_f[c * 3 + j];
    scal[t] = a;
  }
}

// ---------------------------------------------------------------------------
// Kernel 1: WeffT[k][n] = (n<88 && k<229) ? sum_c W_lin[n, c*W + k] * Wv[c] : 0
// Stored transposed + zero padded so WMMA B-fragment loads are branch free.
// ---------------------------------------------------------------------------
__global__ void __launch_bounds__(256)
k_wefft(const float* __restrict__ W_lin, const float* __restrict__ Wv,
        float* __restrict__ WeffT) {
  int idx = blockIdx.x * 256 + threadIdx.x;
  if (idx >= KPAD * NPAD) return;
  int n = idx % NPAD;
  int k = idx / NPAD;
  float acc = 0.f;
  if (n < OUTc && k < Wc) {
    const float* row = W_lin + n * WLIN_STRIDE + k;
#pragma unroll 8
    for (int c = 0; c < Cc; ++c) acc += row[c * Wc] * Wv[c];
  }
  WeffT[idx] = acc;
}

// ---------------------------------------------------------------------------
// Kernel 2: one block per (b,h) row. Stage 3 x (W+2) halo rows in LDS,
// compute 9 energies, softmax, write attn (output 2) and ov (K-padded).
// ---------------------------------------------------------------------------
__global__ void __launch_bounds__(256)
k_attn(const float* __restrict__ spec, const float* __restrict__ scal,
       float* __restrict__ attn_out, float* __restrict__ ov) {
  __shared__ float tile[3][KPAD];   // only cols [0, W+1] used
  const int bh  = blockIdx.x;
  const int b   = bh >> 9;          // H = 512
  const int h   = bh & 511;
  const int tid = threadIdx.x;

  for (int idx = tid; idx < 3 * (Wc + 2); idx += 256) {
    int r  = idx / (Wc + 2);
    int pc = idx % (Wc + 2);
    int gh = h - 1 + r;             // spec row for tile row r
    int gw = pc - 1;                // spec col for padded col pc
    float v = 0.f;
    if ((unsigned)gh < (unsigned)Hc && (unsigned)gw < (unsigned)Wc)
      v = spec[(b * Hc + gh) * Wc + gw];
    tile[r][pc] = v;
  }

  const float alpha = scal[0];
  const float rt0 = scal[1], rt1 = scal[2], rt2 = scal[3];
  const float rf0 = scal[4], rf1 = scal[5], rf2 = scal[6];
  __syncthreads();

  const int w = tid;
  if (w < KPAD) {
    if (w < Wc) {
      const float s = tile[1][w + 1];
      float nbr[9], e[9];
      const float rt[3] = {rt0, rt1, rt2};
      const float rf[3] = {rf0, rf1, rf2};
      float mx = -3.402823466e38f;
#pragma unroll
      for (int i = 0; i < 3; ++i)
#pragma unroll
        for (int j = 0; j < 3; ++j) {
          int k = i * 3 + j;
          nbr[k] = tile[i][w + j];
          e[k]   = s * (alpha * nbr[k] + rt[i] + rf[j]);
          mx     = fmaxf(mx, e[k]);
        }
      float sum = 0.f;
#pragma unroll
      for (int k = 0; k < 9; ++k) { e[k] = __expf(e[k] - mx); sum += e[k]; }
      const float inv = 1.f / sum;
      float ovv = 0.f;
      float* ao = attn_out + (size_t)(bh * Wc + w) * 9;
#pragma unroll
      for (int k = 0; k < 9; ++k) {
        float a = e[k] * inv;
        ao[k] = a;
        ovv += a * nbr[k];
      }
      ov[bh * KPAD + w] = ovv;
    } else {
      ov[bh * KPAD + w] = 0.f;      // zero K padding for the WMMA GEMM
    }
  }
}

// ---------------------------------------------------------------------------
// Kernel 3: frame_pred[m][n] = sigmoid( sum_k ov[m][k]*WeffT[k][n] + b_lin[n] )
// M=2048, N=88 (pad 96), K=229 (pad 232). One wave per 16x16 tile,
// V_WMMA_F32_16X16X4_F32, 58 K-steps.
// A frag (16x4 f32): lane l -> row m0+(l&15), VGPR v -> k = 4*kk + 2*(l>>4)+v
// B frag (4x16 f32): lane l -> col n0+(l&15), VGPR v -> same k
// C/D (16x16 f32):   lane l, VGPR v -> m = v + 8*(l>>4), n = l&15
// ---------------------------------------------------------------------------
__global__ void __launch_bounds__(32)
k_gemm(const float* __restrict__ ov, const float* __restrict__ WeffT,
       const float* __restrict__ b_lin, float* __restrict__ out) {
  const int m0   = blockIdx.x * 16;
  const int n0   = blockIdx.y * 16;
  const int lane = threadIdx.x;
  const int lo   = lane & 15;
  const int hi   = lane >> 4;

  const float* arow = ov + (m0 + lo) * KPAD + 2 * hi;  // 8B aligned
  const float* bcol = WeffT + n0 + lo;

  v8f c = {};
#pragma unroll 4
  for (int kk = 0; kk < KPAD / 4; ++kk) {
    const int k = kk * 4;
    v2f a = *(const v2f*)(arow + k);
    v2f bb;
    bb.x = bcol[(k + 2 * hi)     * NPAD];
    bb.y = bcol[(k + 2 * hi + 1) * NPAD];
    c = __builtin_amdgcn_wmma_f32_16x16x4_f32(
        /*neg_a=*/false, a, /*neg_b=*/false, bb,
        /*c_mod=*/(short)0, c, /*reuse_a=*/false, /*reuse_b=*/false);
  }

  const int n = n0 + lo;
  if (n < OUTc) {
    const float bias = b_lin[n];
#pragma unroll
    for (int v = 0; v < 8; ++v) {
      const int m = m0 + v + 8 * hi;
      const float val = c[v] + bias;
      out[m * OUTc + n] = 1.f / (1.f + __expf(-val));
    }
  }
}

// ---------------------------------------------------------------------------
extern "C" void kernel_launch(void* const* d_in, const int* in_sizes, int n_in,
                              void* d_out, int out_size, void* d_ws, size_t ws_size,
                              hipStream_t stream) {
  const float* spec  = (const float*)d_in[0];
  const float* Wq    = (const float*)d_in[1];
  const float* Wk    = (const float*)d_in[2];
  const float* Wv    = (const float*)d_in[3];
  const float* rel_t = (const float*)d_in[4];
  const float* rel_f = (const float*)d_in[5];
  const float* W_lin = (const float*)d_in[6];
  const float* b_lin = (const float*)d_in[7];

  float* out = (float*)d_out;
  float* ws  = (float*)d_ws;

  k_scalars<<<1, 32, 0, stream>>>(Wq, Wk, rel_t, rel_f, ws + WS_SCAL);

  const int wt = (KPAD * NPAD + 255) / 256;
  k_wefft<<<wt, 256, 0, stream>>>(W_lin, Wv, ws + WS_WEFFT);

  k_attn<<<Mrows, 256, 0, stream>>>(spec, ws + WS_SCAL, out + FP_SIZE, ws + WS_OV);

  dim3 grid(Mrows / 16, NPAD / 16);   // 128 x 6 tiles
  k_gemm<<<grid, 32, 0, stream>>>(ws + WS_OV, ws + WS_WEFFT, b_lin, out);
}